// GraphAutoEncoder_14989435863364
// MI455X (gfx1250) — compile-verified
//
#include <hip/hip_runtime.h>

typedef float v2f __attribute__((ext_vector_type(2)));
typedef float v8f __attribute__((ext_vector_type(8)));

#define THREADS 256
#define NWAVE 8

// ---- packed weight offsets in LDS (floats) ----
#define W1R1 0      // enc_w1 row 1 (64)
#define W1R2 64     // enc_w1 row 2 (64)
#define EB1  128    // enc_b1 (64)
#define EW2  192    // enc_w2 (128)
#define EB2  320    // enc_b2 (2)
#define G1B  322    // gcn1_b (32)
#define G2W  354    // gcn2_w (1024)
#define G2B  1378   // gcn2_b (32)
#define DW1  1410   // dec_w1 (2048)
#define DB1  3458   // dec_b1 (64)
#define DW2  3522   // dec_w2 (192)
#define DB2  3714   // dec_b2 (3)
#define W1P  3720   // gcn1_w zero-padded to 4x32 (128); rows 2..3 = 0
#define WGT_TOTAL 3848

struct alignas(16) WaveScratch {
    float pts[32][2];        // latent points, lane = b*8+s (Gabriel exchange)
    float dinv[32];          // 1/sqrt(deg)
    float latK[2][16][4];    // latent rows padded to K=4 (cols 2,3 = 0)
    float nrm16[2][16][20];  // block-diag 16x16 norm per 2-batch group (pad 20, 16B rows)
    float mat[16][34];       // layout-staging tile (pad 34, keeps 8B align)
    float pool[4][32];       // pooled features per batch in quad
};

static __device__ __forceinline__ v8f wmma4(v2f a, v2f b, v8f c) {
    // D = A(16x4,f32) x B(4x16,f32) + C(16x16,f32)
    return __builtin_amdgcn_wmma_f32_16x16x4_f32(false, a, false, b, (short)0, c, false, false);
}

__global__ __launch_bounds__(THREADS)
void gae_kernel(const float* __restrict__ x,
                const float* __restrict__ enc_w1, const float* __restrict__ enc_b1,
                const float* __restrict__ enc_w2, const float* __restrict__ enc_b2,
                const float* __restrict__ gcn1_w, const float* __restrict__ gcn1_b,
                const float* __restrict__ gcn2_w, const float* __restrict__ gcn2_b,
                const float* __restrict__ dec_w1, const float* __restrict__ dec_b1,
                const float* __restrict__ dec_w2, const float* __restrict__ dec_b2,
                float* __restrict__ outRec, float* __restrict__ outLat, int B)
{
    __shared__ float wgt[WGT_TOTAL];
    __shared__ WaveScratch wsc[NWAVE];

    const int tid = threadIdx.x;
    // ---- preload all weights into LDS ----
    for (int i = tid; i < 64; i += THREADS) {
        wgt[W1R1 + i] = enc_w1[64 + i];
        wgt[W1R2 + i] = enc_w1[128 + i];
        wgt[EB1 + i]  = enc_b1[i];
        wgt[DB1 + i]  = dec_b1[i];
        wgt[W1P + i]      = gcn1_w[i];   // rows 0..1 of padded W1
        wgt[W1P + 64 + i] = 0.0f;        // rows 2..3 zero (K pad)
    }
    for (int i = tid; i < 128;  i += THREADS) wgt[EW2 + i] = enc_w2[i];
    for (int i = tid; i < 32;   i += THREADS) { wgt[G1B + i] = gcn1_b[i]; wgt[G2B + i] = gcn2_b[i]; }
    for (int i = tid; i < 1024; i += THREADS) wgt[G2W + i] = gcn2_w[i];
    for (int i = tid; i < 2048; i += THREADS) wgt[DW1 + i] = dec_w1[i];
    for (int i = tid; i < 192;  i += THREADS) wgt[DW2 + i] = dec_w2[i];
    if (tid < 2) wgt[EB2 + tid] = enc_b2[tid];
    if (tid < 3) wgt[DB2 + tid] = dec_b2[tid];
    __syncthreads();

    const int wave = tid >> 5;
    const int lane = tid & 31;
    WaveScratch& S = wsc[wave];
    const int quad0 = blockIdx.x * 32 + wave * 4;   // first batch of this wave's 4-batch quad
    if (quad0 >= B) return;                         // wave-uniform

    const int b  = lane >> 3;   // batch within quad (0..3)
    const int s  = lane & 7;    // node id
    const float sF = (float)s;
    const bool tailOK = (quad0 + b) < B;
    const int bClamp = tailOK ? (quad0 + b) : (B - 1);

    // ================= encoder (lane = batch x node) =================
    const float xv = x[bClamp * 8 + s];
    float lat0 = wgt[EB2 + 0], lat1 = wgt[EB2 + 1];
#pragma unroll
    for (int j = 0; j < 64; ++j) {
        float h = fmaf(wgt[W1R1 + j], xv, fmaf(wgt[W1R2 + j], sF, wgt[EB1 + j]));
        h = fmaxf(h, 0.0f);
        const float2 w2p = *(const float2*)&wgt[EW2 + 2 * j];
        lat0 = fmaf(h, w2p.x, lat0);
        lat1 = fmaf(h, w2p.y, lat1);
    }
    if (tailOK) {
        float2* o = (float2*)(outLat + (size_t)(bClamp * 8 + s) * 2);
        *o = make_float2(lat0, lat1);
    }
    S.pts[lane][0] = lat0;
    S.pts[lane][1] = lat1;
    {   // latent rows in A-operand-friendly K-padded form (zeros live in LDS, not control flow)
        const int g = b >> 1;
        const int r = ((b & 1) << 3) + s;
        S.latK[g][r][0] = lat0;
        S.latK[g][r][1] = lat1;
        S.latK[g][r][2] = 0.0f;
        S.latK[g][r][3] = 0.0f;
    }

    // ================= Gabriel graph + GCN normalization =================
    float px[8], py[8];
#pragma unroll
    for (int k = 0; k < 8; ++k) { px[k] = S.pts[b * 8 + k][0]; py[k] = S.pts[b * 8 + k][1]; }

    float ah[8];
    float deg = 0.0f;
#pragma unroll
    for (int j = 0; j < 8; ++j) {
        float mx = (px[s] + px[j]) * 0.5f, my = (py[s] + py[j]) * 0.5f;
        float dx = px[s] - mx, dy = py[s] - my;
        float r2 = dx * dx + dy * dy;
        bool blocked = false;
#pragma unroll
        for (int k = 0; k < 8; ++k) {
            float ex = px[k] - mx, ey = py[k] - my;
            bool skip = (k == s) || (k == j);
            blocked = blocked || (!skip && (ex * ex + ey * ey < r2));
        }
        float a = (j == s) ? 1.0f : (blocked ? 0.0f : 1.0f);
        ah[j] = a;
        deg += a;
    }
    const float di = rsqrtf(deg);
    S.dinv[lane] = di;
    float nr[8];
#pragma unroll
    for (int j = 0; j < 8; ++j) nr[j] = ah[j] * di * S.dinv[b * 8 + j];

    {   // scatter row into block-diagonal 16x16 norm; 4 x b128 stores per lane
        const int g  = b >> 1;
        const int r  = ((b & 1) << 3) + s;
        const int bl = b & 1;
#pragma unroll
        for (int c4 = 0; c4 < 4; ++c4) {
            float4 v;
            v.x = (((4 * c4 + 0) >> 3) == bl) ? nr[(4 * c4 + 0) & 7] : 0.0f;
            v.y = (((4 * c4 + 1) >> 3) == bl) ? nr[(4 * c4 + 1) & 7] : 0.0f;
            v.z = (((4 * c4 + 2) >> 3) == bl) ? nr[(4 * c4 + 2) & 7] : 0.0f;
            v.w = (((4 * c4 + 3) >> 3) == bl) ? nr[(4 * c4 + 3) & 7] : 0.0f;
            *(float4*)&S.nrm16[g][r][4 * c4] = v;
        }
    }

    // ================= WMMA GCN pipeline (2 batches per 16-row tile) =================
    const int  lcol   = lane & 15;
    const bool hiHalf = lane >= 16;
    const int  koff   = hiHalf ? 2 : 0;   // K-pair selected by lane half (f32 WMMA layout)
    const int  hi8    = hiHalf ? 8 : 0;   // row offset of C layout for hi lanes

    v8f  h1c[2];
#pragma unroll
    for (int g = 0; g < 2; ++g) {
        // Norm16 as four 16x4 A-tiles
        v2f nA[4];
#pragma unroll
        for (int t = 0; t < 4; ++t) {
            nA[t].x = S.nrm16[g][lcol][4 * t + koff];
            nA[t].y = S.nrm16[g][lcol][4 * t + koff + 1];
        }
        // Latent as 16x4 A operand (zero-padded rows come from LDS -> no EXEC games)
        v2f lA;
        lA.x = S.latK[g][lcol][koff];
        lA.y = S.latK[g][lcol][koff + 1];

#pragma unroll
        for (int nh = 0; nh < 2; ++nh) {
            // W1 B-tile (4x16) from the zero-padded 4x32 LDS image
            v2f w1b;
            w1b.x = wgt[W1P + koff * 32 + 16 * nh + lcol];
            w1b.y = wgt[W1P + (koff + 1) * 32 + 16 * nh + lcol];
            v8f z1 = {};
            z1 = wmma4(lA, w1b, z1);                    // Z1 = Lat @ W1

            // bounce Z1 (C layout) -> B-operand tiles via LDS (same-wave, in-order)
#pragma unroll
            for (int k = 0; k < 8; ++k) S.mat[k + hi8][lcol] = z1[k];

            float bias = wgt[G1B + 16 * nh + lcol];
            v8f c = { bias, bias, bias, bias, bias, bias, bias, bias };
#pragma unroll
            for (int t = 0; t < 4; ++t) {
                v2f bt;
                bt.x = S.mat[4 * t + koff][lcol];
                bt.y = S.mat[4 * t + koff + 1][lcol];
                c = wmma4(nA[t], bt, c);                // H1 = Norm @ Z1 + b1
            }
#pragma unroll
            for (int k = 0; k < 8; ++k) c[k] = fmaxf(c[k], 0.0f);   // relu
            h1c[nh] = c;
        }

        // stage H1 (16x32, C layout) into LDS, reload as A-layout K-tiles
#pragma unroll
        for (int nh = 0; nh < 2; ++nh)
#pragma unroll
            for (int k = 0; k < 8; ++k)
                S.mat[k + hi8][16 * nh + lcol] = h1c[nh][k];

        v2f h1A[8];
#pragma unroll
        for (int t = 0; t < 8; ++t) {
            h1A[t].x = S.mat[lcol][4 * t + koff];
            h1A[t].y = S.mat[lcol][4 * t + koff + 1];
        }

#pragma unroll
        for (int nh = 0; nh < 2; ++nh) {
            v8f z2 = {};
#pragma unroll
            for (int t = 0; t < 8; ++t) {               // Z2 = H1 @ W2, K=32
                v2f wb;
                wb.x = wgt[G2W + (4 * t + koff) * 32 + 16 * nh + lcol];
                wb.y = wgt[G2W + (4 * t + koff + 1) * 32 + 16 * nh + lcol];
                z2 = wmma4(h1A[t], wb, z2);
            }
            // bounce Z2 -> B tiles, then H2 = Norm @ Z2
#pragma unroll
            for (int k = 0; k < 8; ++k) S.mat[k + hi8][lcol] = z2[k];
            v8f h2 = {};
#pragma unroll
            for (int t = 0; t < 4; ++t) {
                v2f bt;
                bt.x = S.mat[4 * t + koff][lcol];
                bt.y = S.mat[4 * t + koff + 1][lcol];
                h2 = wmma4(nA[t], bt, h2);
            }
            // pooled = mean over 8 nodes + gcn2 bias (bias constant over nodes)
            float ps = h2[0] + h2[1] + h2[2] + h2[3] + h2[4] + h2[5] + h2[6] + h2[7];
            ps = ps * 0.125f + wgt[G2B + 16 * nh + lcol];
            S.pool[2 * g + (hiHalf ? 1 : 0)][16 * nh + lcol] = ps;
        }
    }

    // ================= decoder (8 lanes per batch, only dec column 1 needed) ====
    float pl[32];
#pragma unroll
    for (int f = 0; f < 32; ++f) pl[f] = S.pool[b][f];

    float acc = 0.0f;
#pragma unroll
    for (int j0 = 0; j0 < 8; ++j0) {
        const int j = j0 * 8 + s;
        float d = wgt[DB1 + j];
#pragma unroll
        for (int f = 0; f < 32; ++f) d = fmaf(pl[f], wgt[DW1 + f * 64 + j], d);
        d = fmaxf(d, 0.0f);
        acc = fmaf(d, wgt[DW2 + j * 3 + 1], acc);
    }
    acc += __shfl_xor(acc, 1);
    acc += __shfl_xor(acc, 2);
    acc += __shfl_xor(acc, 4);
    const float rec = acc + wgt[DB2 + 1];
    if (tailOK) outRec[bClamp * 8 + s] = rec;
}

extern "C" void kernel_launch(void* const* d_in, const int* in_sizes, int n_in,
                              void* d_out, int out_size, void* d_ws, size_t ws_size,
                              hipStream_t stream) {
    (void)n_in; (void)out_size; (void)d_ws; (void)ws_size;
    const float* x      = (const float*)d_in[0];
    const float* enc_w1 = (const float*)d_in[1];
    const float* enc_b1 = (const float*)d_in[2];
    const float* enc_w2 = (const float*)d_in[3];
    const float* enc_b2 = (const float*)d_in[4];
    const float* gcn1_w = (const float*)d_in[5];
    const float* gcn1_b = (const float*)d_in[6];
    const float* gcn2_w = (const float*)d_in[7];
    const float* gcn2_b = (const float*)d_in[8];
    const float* dec_w1 = (const float*)d_in[9];
    const float* dec_b1 = (const float*)d_in[10];
    const float* dec_w2 = (const float*)d_in[11];
    const float* dec_b2 = (const float*)d_in[12];

    const int B = in_sizes[0] / 8;
    float* outRec = (float*)d_out;
    float* outLat = outRec + (size_t)B * 8;

    const int batchesPerBlock = 32;                 // 8 waves x 4 batches
    dim3 grid((B + batchesPerBlock - 1) / batchesPerBlock);
    dim3 block(THREADS);
    hipLaunchKernelGGL(gae_kernel, grid, block, 0, stream,
                       x, enc_w1, enc_b1, enc_w2, enc_b2,
                       gcn1_w, gcn1_b, gcn2_w, gcn2_b,
                       dec_w1, dec_b1, dec_w2, dec_b2,
                       outRec, outLat, B);
}